// DecNP_6012954214675
// MI455X (gfx1250) — compile-verified
//
#include <hip/hip_runtime.h>
#include <hip/hip_bf16.h>

typedef float v2f __attribute__((ext_vector_type(2)));
typedef float v8f __attribute__((ext_vector_type(8)));

#define FP_BIG 3.0e38f

// Branchless insert of (s, ci) into ascending top-3 (d0<=d1<=d2).
// 3 compares + 10 selects, straight-line: no EXEC-mask branching.
__device__ __forceinline__ void ins3(float s, int ci,
                                     float& d0, int& i0,
                                     float& d1, int& i1,
                                     float& d2, int& i2) {
    const bool c2 = s < d2;
    const bool c1 = s < d1;
    const bool c0 = s < d0;
    d2 = c1 ? d1 : (c2 ? s  : d2);
    i2 = c1 ? i1 : (c2 ? ci : i2);
    d1 = c0 ? d0 : (c1 ? s  : d1);
    i1 = c0 ? i0 : (c1 ? ci : i1);
    d0 = c0 ? s  : d0;
    i0 = c0 ? ci : i0;
}

// One wave32 per 16 fine points. Distance-score tiles (16 coarse x 16 fine)
// via V_WMMA_F32_16X16X4_F32:
//   A (16x4, coarse rows) = [-2x, -2y | -2z, |c|^2]  (lanes 0-15: K0,K1; 16-31: K2,K3)
//   B (4x16, fine cols)   = [ x,   y  |  z,   1   ]
//   D[m][n] = |c_m|^2 - 2 f_n.c_m  (monotone in true distance per fine point n)
// C/D layout: lane n holds rows 0-7 (v8f), lane n+16 holds rows 8-15.
// Coarse loop unrolled x2 so two independent WMMAs cover the WMMA->VALU hazard.
__global__ __launch_bounds__(128) void cdna5_topk3(
    const float* __restrict__ fine,    // [B, N, 3]
    const float* __restrict__ coarse,  // [B, S, 3]
    int*   __restrict__ oidx,          // [B, N, 3]
    float* __restrict__ ow,            // [B, N, 3]
    int N, int S)
{
    const int lane = threadIdx.x & 31;
    const int wave = threadIdx.x >> 5;
    const int tile = blockIdx.x * 4 + wave;
    const int tilesPerBatch = N >> 4;
    const int b   = tile / tilesPerBatch;
    const int t   = tile - b * tilesPerBatch;
    const int col = lane & 15;   // fine point within tile
    const int hi  = lane >> 4;   // 0: K0/K1 + rows 0-7, 1: K2/K3 + rows 8-15
    const int n   = t * 16 + col;

    const float* fp = fine + ((size_t)b * N + n) * 3;
    const float fx = fp[0], fy = fp[1], fz = fp[2];

    v2f bm;
    bm.x = hi ? fz : fx;
    bm.y = hi ? 1.0f : fy;

    float d0 = FP_BIG, d1 = FP_BIG, d2 = FP_BIG;
    int   i0 = 0, i1 = 0, i2 = 0;

    const float* cbase = coarse + (size_t)b * S * 3;

    for (int s0 = 0; s0 < S; s0 += 32) {
        const float* cpA = cbase + (size_t)(s0 + col) * 3;
        const float* cpB = cbase + (size_t)(s0 + 16 + col) * 3;
        const float cxA = cpA[0], cyA = cpA[1], czA = cpA[2];
        const float cxB = cpB[0], cyB = cpB[1], czB = cpB[2];
        const float cnA = cxA * cxA + cyA * cyA + czA * czA;
        const float cnB = cxB * cxB + cyB * cyB + czB * czB;

        v2f amA, amB;
        amA.x = hi ? (-2.0f * czA) : (-2.0f * cxA);
        amA.y = hi ? cnA           : (-2.0f * cyA);
        amB.x = hi ? (-2.0f * czB) : (-2.0f * cxB);
        amB.y = hi ? cnB           : (-2.0f * cyB);

        v8f accA = {0.f, 0.f, 0.f, 0.f, 0.f, 0.f, 0.f, 0.f};
        v8f accB = {0.f, 0.f, 0.f, 0.f, 0.f, 0.f, 0.f, 0.f};
        accA = __builtin_amdgcn_wmma_f32_16x16x4_f32(
            false, amA, false, bm, (short)0, accA, false, false);
        accB = __builtin_amdgcn_wmma_f32_16x16x4_f32(
            false, amB, false, bm, (short)0, accB, false, false);

#pragma unroll
        for (int v = 0; v < 8; ++v) {
            ins3(accA[v], s0 + v + hi * 8,      d0, i0, d1, i1, d2, i2);
        }
#pragma unroll
        for (int v = 0; v < 8; ++v) {
            ins3(accB[v], s0 + 16 + v + hi * 8, d0, i0, d1, i1, d2, i2);
        }
    }

    // Merge the two half-row top-3 sets for the same fine point (lane ^ 16).
    {
        float e0 = __shfl_xor(d0, 16, 32);
        float e1 = __shfl_xor(d1, 16, 32);
        float e2 = __shfl_xor(d2, 16, 32);
        int   j0 = __shfl_xor(i0, 16, 32);
        int   j1 = __shfl_xor(i1, 16, 32);
        int   j2 = __shfl_xor(i2, 16, 32);
        ins3(e0, j0, d0, i0, d1, i1, d2, i2);
        ins3(e1, j1, d0, i0, d1, i1, d2, i2);
        ins3(e2, j2, d0, i0, d1, i1, d2, i2);
    }

    if (hi == 0) {
        // Recompute exact squared distances for the 3 winners -> weights.
        const float* c0 = cbase + (size_t)i0 * 3;
        const float* c1 = cbase + (size_t)i1 * 3;
        const float* c2 = cbase + (size_t)i2 * 3;
        float ax = fx - c0[0], ay = fy - c0[1], az = fz - c0[2];
        float bx = fx - c1[0], by = fy - c1[1], bz = fz - c1[2];
        float gx = fx - c2[0], gy = fy - c2[1], gz = fz - c2[2];
        float q0 = ax * ax + ay * ay + az * az;
        float q1 = bx * bx + by * by + bz * bz;
        float q2 = gx * gx + gy * gy + gz * gz;
        float r0 = 1.0f / (q0 + 1e-8f);
        float r1 = 1.0f / (q1 + 1e-8f);
        float r2 = 1.0f / (q2 + 1e-8f);
        float rs = 1.0f / (r0 + r1 + r2);
        const size_t o = ((size_t)b * N + n) * 3;
        oidx[o + 0] = i0; oidx[o + 1] = i1; oidx[o + 2] = i2;
        ow[o + 0] = r0 * rs; ow[o + 1] = r1 * rs; ow[o + 2] = r2 * rs;
    }
}

// Stage-1 interpolation: out[b][c][n] = sum_k w[b][n][k] * src[b][c][idx[b][n][k]]
__global__ __launch_bounds__(256) void cdna5_interp(
    const float* __restrict__ src,   // [B, D, S]
    const int*   __restrict__ idx,   // [B, N, 3]
    const float* __restrict__ wgt,   // [B, N, 3]
    float* __restrict__ out,         // [B, D, N]
    int N, int S, int D)
{
    const int b = blockIdx.y;
    const int c = blockIdx.x;
    const float* row  = src + ((size_t)b * D + c) * S;
    float*       orow = out + ((size_t)b * D + c) * N;
    const int*   ib   = idx + (size_t)b * N * 3;
    const float* wb   = wgt + (size_t)b * N * 3;
    for (int n = threadIdx.x; n < N; n += 256) {
        const int o = n * 3;
        orow[n] = wb[o + 0] * row[ib[o + 0]]
                + wb[o + 1] * row[ib[o + 1]]
                + wb[o + 2] * row[ib[o + 2]];
    }
}

// Stage-2 interpolation + both concats fused into the final [B,896,8192] output.
// Channels: [0,128) copy of x0; [128,384) gather from x1; [384,896) gather from interp1.
__global__ __launch_bounds__(256) void cdna5_final(
    const float* __restrict__ x0,       // [B, 128, N]
    const float* __restrict__ x1,       // [B, 256, S]
    const float* __restrict__ interp1,  // [B, 512, S]
    const int*   __restrict__ idx,      // [B, N, 3]
    const float* __restrict__ wgt,      // [B, N, 3]
    float* __restrict__ out,            // [B, 896, N]
    int N, int S)
{
    const int b = blockIdx.y;
    const int c = blockIdx.x;   // 0..895
    float* orow = out + ((size_t)b * 896 + c) * N;

    if (c < 128) {
        const float* xr = x0 + ((size_t)b * 128 + c) * N;
        for (int n = threadIdx.x; n < N; n += 256) orow[n] = xr[n];
        return;
    }

    const float* row = (c < 384)
        ? x1      + ((size_t)b * 256 + (c - 128)) * S
        : interp1 + ((size_t)b * 512 + (c - 384)) * S;
    const int*   ib = idx + (size_t)b * N * 3;
    const float* wb = wgt + (size_t)b * N * 3;
    for (int n = threadIdx.x; n < N; n += 256) {
        const int o = n * 3;
        orow[n] = wb[o + 0] * row[ib[o + 0]]
                + wb[o + 1] * row[ib[o + 1]]
                + wb[o + 2] * row[ib[o + 2]];
    }
}

extern "C" void kernel_launch(void* const* d_in, const int* in_sizes, int n_in,
                              void* d_out, int out_size, void* d_ws, size_t ws_size,
                              hipStream_t stream) {
    (void)in_sizes; (void)n_in; (void)out_size; (void)ws_size;
    const float* xyz0 = (const float*)d_in[0];  // [4,8192,3]
    const float* xyz1 = (const float*)d_in[1];  // [4,2048,3]
    const float* xyz2 = (const float*)d_in[2];  // [4,512,3]
    const float* x0   = (const float*)d_in[3];  // [4,128,8192]
    const float* x1   = (const float*)d_in[4];  // [4,256,2048]
    const float* x2   = (const float*)d_in[5];  // [4,512,512]
    float* out = (float*)d_out;                 // [4,896,8192]

    const int B = 4;
    const int N0 = 8192, N1 = 2048, N2 = 512;

    // Workspace layout (~17.8 MB)
    float* interp1 = (float*)d_ws;                         // [4,512,2048]
    int*   idx1    = (int*)(interp1 + (size_t)B * 512 * N1);
    float* w1      = (float*)(idx1 + (size_t)B * N1 * 3);
    int*   idx2    = (int*)(w1 + (size_t)B * N1 * 3);
    float* w2      = (float*)(idx2 + (size_t)B * N0 * 3);

    // Stage 1: top-3 of 2048 fine vs 512 coarse. 4*2048/16 = 512 waves.
    cdna5_topk3<<<(B * N1 / 16) / 4, 128, 0, stream>>>(xyz1, xyz2, idx1, w1, N1, N2);

    // Stage 1: interpolate x2 (512 channels) onto the 2048-point level.
    cdna5_interp<<<dim3(512, B), 256, 0, stream>>>(x2, idx1, w1, interp1, N1, N2, 512);

    // Stage 2: top-3 of 8192 fine vs 2048 coarse. 4*8192/16 = 2048 waves.
    cdna5_topk3<<<(B * N0 / 16) / 4, 128, 0, stream>>>(xyz0, xyz1, idx2, w2, N0, N1);

    // Stage 2: fused gather of [x1 | interp1] + concat with x0 -> [4,896,8192].
    cdna5_final<<<dim3(896, B), 256, 0, stream>>>(x0, x1, interp1, idx2, w2, out, N0, N1);
}